// SparseRNN_18519898980708
// MI455X (gfx1250) — compile-verified
//
#include <hip/hip_runtime.h>
#include <cstdint>
#include <cstddef>

namespace {
constexpr int kH   = 50000;   // hidden size (rows of both matrices)
constexpr int kIn  = 50000;   // input size
constexpr int kNnz = 800000;
constexpr int kB   = 8;
constexpr int kT   = 32;
constexpr int kTB  = kT * kB; // 256 dense columns for the ih pass
}

#if defined(__gfx1250__) && __has_builtin(__builtin_amdgcn_global_load_async_to_lds_b32) && \
    __has_builtin(__builtin_amdgcn_s_wait_asynccnt)
#define USE_ASYNC_LDS 1
typedef __attribute__((address_space(1))) int* gbl_i32_ptr;
typedef __attribute__((address_space(3))) int* lds_i32_ptr;
#else
#define USE_ASYNC_LDS 0
#endif

// ---------------- CSR build: zero / count / scan / scatter ----------------

__global__ void zero_i32_kernel(int* __restrict__ p, int n) {
  int i = blockIdx.x * blockDim.x + threadIdx.x;
  if (i < n) p[i] = 0;
}

__global__ void count_rows_kernel(const int* __restrict__ rows, int nnz,
                                  int* __restrict__ rowStart) {
  int i = blockIdx.x * blockDim.x + threadIdx.x;
  if (i < nnz) atomicAdd(&rowStart[rows[i] + 1], 1);
}

// Single-block prefix sum over rowStart[1..n] (counts -> CSR offsets).
__global__ __launch_bounds__(1024) void scan_rowstart_kernel(int* __restrict__ rowStart, int n) {
  __shared__ int sbuf[1024];
  __shared__ int scarry;
  const int tid = threadIdx.x;
  if (tid == 0) scarry = 0;
  __syncthreads();
  constexpr int PER = 4;
  constexpr int CH  = 1024 * PER;
  for (int base = 0; base < n; base += CH) {
    int v[PER];
    int tot  = 0;
    int idx0 = base + tid * PER;
#pragma unroll
    for (int k = 0; k < PER; ++k) {
      int idx = idx0 + k;
      v[k] = (idx < n) ? rowStart[idx + 1] : 0;
      tot += v[k];
    }
    sbuf[tid] = tot;
    __syncthreads();
    for (int off = 1; off < 1024; off <<= 1) {
      int add = (tid >= off) ? sbuf[tid - off] : 0;
      __syncthreads();
      sbuf[tid] += add;
      __syncthreads();
    }
    int run = sbuf[tid] - tot + scarry;   // exclusive offset for this thread
#pragma unroll
    for (int k = 0; k < PER; ++k) {
      int idx = idx0 + k;
      run += v[k];
      if (idx < n) rowStart[idx + 1] = run;
    }
    int blockTot = sbuf[1023];
    __syncthreads();                      // everyone done reading scarry/sbuf
    if (tid == 0) scarry += blockTot;
    __syncthreads();
  }
}

__global__ void scatter_csr_kernel(const int* __restrict__ rows, const int* __restrict__ cols,
                                   const float* __restrict__ vals, int nnz,
                                   const int* __restrict__ rowStart, int* __restrict__ cursor,
                                   int* __restrict__ colP, float* __restrict__ valP) {
  int i = blockIdx.x * blockDim.x + threadIdx.x;
  if (i >= nnz) return;
  int r   = rows[i];
  int pos = rowStart[r] + atomicAdd(&cursor[r], 1);
  colP[pos] = cols[i];
  valP[pos] = vals[i];
}

// ---------------- x (B,T,IN) -> xT (IN, T*B) so each column c is 1KB contiguous

__global__ void transpose_x_kernel(const float* __restrict__ x, float* __restrict__ xT) {
  int i = blockIdx.x * blockDim.x + threadIdx.x;  // over kT*kIn, c fastest
  if (i >= kT * kIn) return;
  int t = i / kIn;
  int c = i - t * kIn;
  float v[kB];
#pragma unroll
  for (int b = 0; b < kB; ++b)
    v[b] = x[(size_t)b * (kT * kIn) + (size_t)t * kIn + c];
  float4* dst = reinterpret_cast<float4*>(xT + (size_t)c * kTB + (size_t)t * kB);
  dst[0] = make_float4(v[0], v[1], v[2], v[3]);
  dst[1] = make_float4(v[4], v[5], v[6], v[7]);
}

// ---------------- ih pass: all timesteps at once, block-per-row, no atomics.
// Async double-buffered LDS staging of the 1KB xT row per nnz (gfx1250 path).

__global__ __launch_bounds__(256) void ih_rows_kernel(const int* __restrict__ rowStart,
                                                      const int* __restrict__ colP,
                                                      const float* __restrict__ valP,
                                                      const float* __restrict__ xT,
                                                      float* __restrict__ out) {
  const int r   = blockIdx.x;         // row of W_ih
  const int tid = threadIdx.x;        // 0..255 == t*8+b
  const int s   = rowStart[r];
  const int n   = rowStart[r + 1] - s;
  float acc = 0.f;
#if USE_ASYNC_LDS
  __shared__ float buf[2][kTB];
  if (n > 0) {
    const float* g0 = xT + (size_t)colP[s] * kTB + tid;
    __builtin_amdgcn_global_load_async_to_lds_b32(
        (gbl_i32_ptr)(uintptr_t)g0,
        (lds_i32_ptr)(uint32_t)(uintptr_t)&buf[0][tid], 0, 0);
  }
  for (int jj = 0; jj < n; ++jj) {
    const int cur = jj & 1;
    if (jj + 1 < n) {
      const float* g1 = xT + (size_t)colP[s + jj + 1] * kTB + tid;
      __builtin_amdgcn_global_load_async_to_lds_b32(
          (gbl_i32_ptr)(uintptr_t)g1,
          (lds_i32_ptr)(uint32_t)(uintptr_t)&buf[cur ^ 1][tid], 0, 0);
      __builtin_amdgcn_s_wait_asynccnt(1);   // current buffer complete (in-order)
    } else {
      __builtin_amdgcn_s_wait_asynccnt(0);
    }
    acc = fmaf(valP[s + jj], buf[cur][tid], acc);  // per-lane: reads its own async write
  }
#else
  for (int jj = 0; jj < n; ++jj)
    acc = fmaf(valP[s + jj], xT[(size_t)colP[s + jj] * kTB + tid], acc);
#endif
  const int t = tid >> 3, b = tid & 7;
  out[(size_t)b * (kT * kH) + (size_t)t * kH + r] = acc;  // out doubles as preIH storage
}

// ---------------- recurrence: hh gather (no atomics) + fused bias/tanh ------

__global__ __launch_bounds__(256) void hh_step_kernel(const int* __restrict__ rowStart,
                                                      const int* __restrict__ colP,
                                                      const float* __restrict__ valP,
                                                      const float* __restrict__ h,
                                                      float* __restrict__ pre) {
  int i = blockIdx.x * blockDim.x + threadIdx.x;   // over kH*kB
  if (i >= kH * kB) return;
  int r = i >> 3, b = i & 7;                       // 8 lanes share a row -> scalar col/val
  int s = rowStart[r], e = rowStart[r + 1];
  float acc = 0.f;
  for (int j = s; j < e; ++j)
    acc = fmaf(valP[j], h[(size_t)colP[j] * kB + b], acc);   // 32B coalesced across b
  pre[i] = acc;                                    // overwrite: no init kernel needed
}

__global__ __launch_bounds__(256) void combine_kernel(const float* __restrict__ pre,
                                                      const float* __restrict__ bias,
                                                      float* __restrict__ out,
                                                      float* __restrict__ h,
                                                      int t, int use_pre) {
  int i = blockIdx.x * blockDim.x + threadIdx.x;   // over kH*kB, r fastest for coalesced out
  if (i >= kH * kB) return;
  int b = i / kH, r = i - b * kH;
  size_t oidx = (size_t)b * (kT * kH) + (size_t)t * kH + r;
  float v = bias[r] + out[oidx];                   // out currently holds ih contribution
  if (use_pre) v += pre[(size_t)r * kB + b];
  float hv = tanhf(v);
  h[(size_t)r * kB + b] = hv;
  out[oidx] = hv;
}

// ---------------------------------------------------------------------------

extern "C" void kernel_launch(void* const* d_in, const int* in_sizes, int n_in,
                              void* d_out, int out_size, void* d_ws, size_t ws_size,
                              hipStream_t stream) {
  (void)in_sizes; (void)n_in; (void)out_size;
  const float* x      = (const float*)d_in[0];
  const float* hhVals = (const float*)d_in[1];
  const float* hhBias = (const float*)d_in[2];
  const float* ihVals = (const float*)d_in[3];
  const int*   hhIdx  = (const int*)d_in[4];   // (2, NNZ): rows then cols
  const int*   ihIdx  = (const int*)d_in[5];
  float* out = (float*)d_out;

  char*  ws  = (char*)d_ws;
  size_t off = 0;
  auto alloc = [&](size_t bytes) -> void* {
    void* p = ws + off;
    off = (off + bytes + 255) & ~(size_t)255;
    return p;
  };
  float* xT     = (float*)alloc((size_t)kIn * kTB * sizeof(float));   // 51.2 MB
  float* h      = (float*)alloc((size_t)kH * kB * sizeof(float));
  float* pre    = (float*)alloc((size_t)kH * kB * sizeof(float));
  int*   meta   = (int*)  alloc((size_t)(4 * kH + 2) * sizeof(int));  // zeroed block
  int*   colPhh = (int*)  alloc((size_t)kNnz * sizeof(int));
  float* valPhh = (float*)alloc((size_t)kNnz * sizeof(float));
  int*   colPih = (int*)  alloc((size_t)kNnz * sizeof(int));
  float* valPih = (float*)alloc((size_t)kNnz * sizeof(float));
  if (off > ws_size) return;  // workspace too small; bail deterministically

  int* rsHH  = meta;                 // kH+1
  int* curHH = meta + (kH + 1);      // kH
  int* rsIH  = meta + (2 * kH + 1);  // kH+1
  int* curIH = meta + (3 * kH + 2);  // kH

  const int TPB      = 256;
  const int gridNnz  = (kNnz + TPB - 1) / TPB;
  const int gridMeta = (4 * kH + 2 + TPB - 1) / TPB;
  const int gridHB   = (kH * kB + TPB - 1) / TPB;

  // One-time (per launch) CSR build + x transpose.
  zero_i32_kernel<<<gridMeta, TPB, 0, stream>>>(meta, 4 * kH + 2);
  count_rows_kernel<<<gridNnz, TPB, 0, stream>>>(hhIdx, kNnz, rsHH);
  count_rows_kernel<<<gridNnz, TPB, 0, stream>>>(ihIdx, kNnz, rsIH);
  scan_rowstart_kernel<<<1, 1024, 0, stream>>>(rsHH, kH);
  scan_rowstart_kernel<<<1, 1024, 0, stream>>>(rsIH, kH);
  scatter_csr_kernel<<<gridNnz, TPB, 0, stream>>>(hhIdx, hhIdx + kNnz, hhVals, kNnz,
                                                  rsHH, curHH, colPhh, valPhh);
  scatter_csr_kernel<<<gridNnz, TPB, 0, stream>>>(ihIdx, ihIdx + kNnz, ihVals, kNnz,
                                                  rsIH, curIH, colPih, valPih);
  transpose_x_kernel<<<(kT * kIn + TPB - 1) / TPB, TPB, 0, stream>>>(x, xT);

  // ih contribution for ALL timesteps up front, written straight into d_out.
  ih_rows_kernel<<<kH, 256, 0, stream>>>(rsIH, colPih, valPih, xT, out);

  // Sequential recurrence: only the small hh gather + fused tanh per step.
  for (int t = 0; t < kT; ++t) {
    if (t > 0) hh_step_kernel<<<gridHB, TPB, 0, stream>>>(rsHH, colPhh, valPhh, h, pre);
    combine_kernel<<<gridHB, TPB, 0, stream>>>(pre, hhBias, out, h, t, t > 0 ? 1 : 0);
  }
}